// GCN6_80917183857362
// MI455X (gfx1250) — compile-verified
//
#include <hip/hip_runtime.h>

typedef __attribute__((ext_vector_type(2))) float v2f;
typedef __attribute__((ext_vector_type(8))) float v8f;

#define N_NODES 100000
#define DOUT    64

// Guaranteed single-instruction HW float atomic (no-return form, STOREcnt).
// CDNA5 VGLOBAL opcode 86: GLOBAL_ATOMIC_ADD_F32 (RMW executes in L2 atomic units).
__device__ __forceinline__ void atomic_add_f32_hw(float* p, float v) {
    asm volatile("global_atomic_add_f32 %0, %1, off" : : "v"(p), "v"(v) : "memory");
}

// ---------------------------------------------------------------------------
// Degree / normalization kernels
// ---------------------------------------------------------------------------
__global__ void k_fill1(float* __restrict__ p, int n) {
    int i = blockIdx.x * blockDim.x + threadIdx.x;
    if (i < n) p[i] = 1.0f;   // self-loop contributes 1 to every node's degree
}

__global__ void k_degcount(const long long* __restrict__ dst, float* deg, int E) {
    int i = blockIdx.x * blockDim.x + threadIdx.x;
    if (i < E) atomic_add_f32_hw(&deg[(int)dst[i]], 1.0f);
}

__global__ void k_rsqrt_inplace(float* __restrict__ p, int n) {
    int i = blockIdx.x * blockDim.x + threadIdx.x;
    if (i < n) p[i] = rsqrtf(p[i]);   // deg >= 1 always (self-loops)
}

// ---------------------------------------------------------------------------
// Dense GEMM: T[N,64] = X[N,DIN] @ W[DIN,64]  via V_WMMA_F32_16X16X4_F32
// One wave handles a 16-row M tile x full 64-col output (4 N-tiles of 16).
// 32-bit A 16x4 layout: lanes 0-15 hold K={0,1}, lanes 16-31 hold K={2,3};
// fragment element v corresponds to k = kbase + 2*(lane>>4) + v for both A & B.
// C/D layout: VGPR r holds M = r + 8*(lane>>4), N = lane&15.
// ---------------------------------------------------------------------------
template <int DIN>
__global__ __launch_bounds__(256) void k_gemm_wmma(const float* __restrict__ X,
                                                   const float* __restrict__ W,
                                                   float* __restrict__ T,
                                                   int nrows) {
    const int wave = (blockIdx.x * blockDim.x + threadIdx.x) >> 5;
    const int lane = threadIdx.x & 31;
    const int m0 = wave * 16;
    if (m0 >= nrows) return;              // wave-uniform: EXEC stays all-ones

    const int r  = lane & 15;             // row (A) / col (B,C,D) within tile
    const int kg = lane >> 4;             // K half-group

    v8f acc0 = {}, acc1 = {}, acc2 = {}, acc3 = {};
    const float* xrow = X + (size_t)(m0 + r) * DIN;

    for (int kb = 0; kb < DIN; kb += 4) {
        const int k0 = kb + 2 * kg;
        v2f a;
        a.x = xrow[k0];
        a.y = xrow[k0 + 1];
        const float* w0 = W + (size_t)k0 * DOUT;        // row k0 of W
        const float* w1 = w0 + DOUT;                    // row k0+1 of W
        v2f b0, b1, b2, b3;
        b0.x = w0[r];      b0.y = w1[r];
        b1.x = w0[16 + r]; b1.y = w1[16 + r];
        b2.x = w0[32 + r]; b2.y = w1[32 + r];
        b3.x = w0[48 + r]; b3.y = w1[48 + r];
        acc0 = __builtin_amdgcn_wmma_f32_16x16x4_f32(false, a, false, b0, (short)0, acc0, false, false);
        acc1 = __builtin_amdgcn_wmma_f32_16x16x4_f32(false, a, false, b1, (short)0, acc1, false, false);
        acc2 = __builtin_amdgcn_wmma_f32_16x16x4_f32(false, a, false, b2, (short)0, acc2, false, false);
        acc3 = __builtin_amdgcn_wmma_f32_16x16x4_f32(false, a, false, b3, (short)0, acc3, false, false);
    }

    float* trow = T + (size_t)m0 * DOUT;
#pragma unroll
    for (int rr = 0; rr < 8; ++rr) {
        const int m = rr + 8 * kg;
        trow[(size_t)m * DOUT + r]      = acc0[rr];
        trow[(size_t)m * DOUT + 16 + r] = acc1[rr];
        trow[(size_t)m * DOUT + 32 + r] = acc2[rr];
        trow[(size_t)m * DOUT + 48 + r] = acc3[rr];
    }
}

// ---------------------------------------------------------------------------
// agg[i,:] = T[i,:] * dinv[i]^2   (self-loop term; also initializes agg)
// 16 threads per row, float4 each.
// ---------------------------------------------------------------------------
__global__ void k_selfloop_init(const float* __restrict__ T,
                                const float* __restrict__ dinv,
                                float* __restrict__ agg, int n) {
    int tid = blockIdx.x * blockDim.x + threadIdx.x;
    int row = tid >> 4;
    if (row >= n) return;
    int g = (tid & 15) * 4;
    float w = dinv[row];
    w = w * w;
    const float4 v = *(const float4*)(T + (size_t)row * DOUT + g);
    float4 o;
    o.x = v.x * w; o.y = v.y * w; o.z = v.z * w; o.w = v.w * w;
    *(float4*)(agg + (size_t)row * DOUT + g) = o;
}

// ---------------------------------------------------------------------------
// Edge scatter: agg[dst,:] += T[src,:] * dinv[src]*dinv[dst]
// 16 threads per edge, float4 (b128) gather + 4 HW f32 atomics each.
// Feature table + accumulator are L2-resident (2 x 25.6 MB << 192 MB L2).
// ---------------------------------------------------------------------------
__global__ void k_edge_scatter(const long long* __restrict__ src,
                               const long long* __restrict__ dst,
                               const float* __restrict__ dinv,
                               const float* __restrict__ T,
                               float* __restrict__ agg, int E) {
    int tid = blockIdx.x * blockDim.x + threadIdx.x;
    int e = tid >> 4;
    if (e >= E) return;
    int g = (tid & 15) * 4;
    const int s = (int)src[e];
    const int d = (int)dst[e];
    const float nrm = dinv[s] * dinv[d];
    const float4 v = *(const float4*)(T + (size_t)s * DOUT + g);
    float* ap = agg + (size_t)d * DOUT + g;
    atomic_add_f32_hw(ap + 0, v.x * nrm);
    atomic_add_f32_hw(ap + 1, v.y * nrm);
    atomic_add_f32_hw(ap + 2, v.z * nrm);
    atomic_add_f32_hw(ap + 3, v.w * nrm);
}

// ---------------------------------------------------------------------------
// h = relu(h + b), in place. 16 threads per row, float4 each.
// ---------------------------------------------------------------------------
__global__ void k_bias_relu(float* __restrict__ h, const float* __restrict__ b, int n) {
    int tid = blockIdx.x * blockDim.x + threadIdx.x;
    int row = tid >> 4;
    if (row >= n) return;
    int g = (tid & 15) * 4;
    float4 v = *(float4*)(h + (size_t)row * DOUT + g);
    v.x = fmaxf(v.x + b[g + 0], 0.0f);
    v.y = fmaxf(v.y + b[g + 1], 0.0f);
    v.z = fmaxf(v.z + b[g + 2], 0.0f);
    v.w = fmaxf(v.w + b[g + 3], 0.0f);
    *(float4*)(h + (size_t)row * DOUT + g) = v;
}

// ---------------------------------------------------------------------------
// out[i,:] = log_softmax(h[i,:]) over 64 cols; one wave32 per row.
// ---------------------------------------------------------------------------
__global__ void k_log_softmax(const float* __restrict__ h, float* __restrict__ out, int n) {
    int wave = (blockIdx.x * blockDim.x + threadIdx.x) >> 5;
    int lane = threadIdx.x & 31;
    if (wave >= n) return;
    const float* row = h + (size_t)wave * DOUT;
    float a = row[lane];
    float b = row[lane + 32];
    float m = fmaxf(a, b);
#pragma unroll
    for (int o = 16; o > 0; o >>= 1) m = fmaxf(m, __shfl_xor(m, o, 32));
    float s = __expf(a - m) + __expf(b - m);
#pragma unroll
    for (int o = 16; o > 0; o >>= 1) s += __shfl_xor(s, o, 32);
    const float l = m + __logf(s);
    out[(size_t)wave * DOUT + lane]      = a - l;
    out[(size_t)wave * DOUT + lane + 32] = b - l;
}

// ---------------------------------------------------------------------------
// Host-side orchestration
// ---------------------------------------------------------------------------
extern "C" void kernel_launch(void* const* d_in, const int* in_sizes, int n_in,
                              void* d_out, int out_size, void* d_ws, size_t ws_size,
                              hipStream_t stream) {
    const float*     x  = (const float*)d_in[0];
    const long long* ei = (const long long*)d_in[1];   // int64 [2, E]
    const int E = in_sizes[1] / 2;                      // 1,600,000
    const int N = N_NODES;

    const float* Ws[6] = {(const float*)d_in[2], (const float*)d_in[4],
                          (const float*)d_in[6], (const float*)d_in[8],
                          (const float*)d_in[10], (const float*)d_in[12]};
    const float* bs[6] = {(const float*)d_in[3], (const float*)d_in[5],
                          (const float*)d_in[7], (const float*)d_in[9],
                          (const float*)d_in[11], (const float*)d_in[13]};

    // Workspace layout (bytes):  dinv | T | H0 | H1
    const size_t FEAT = (size_t)N * DOUT * sizeof(float);   // 25.6 MB
    char* ws = (char*)d_ws;
    float* dinv = (float*)ws;                               // N floats
    float* T    = (float*)(ws + 512 * 1024);
    float* H0   = (float*)(ws + 512 * 1024 + FEAT);
    float* H1   = (float*)(ws + 512 * 1024 + 2 * FEAT);

    const long long* srcp = ei;       // edge_index[0]
    const long long* dstp = ei + E;   // edge_index[1]

    const int B = 256;
    // --- degree -> dinv = rsqrt(deg) ---
    k_fill1<<<(N + B - 1) / B, B, 0, stream>>>(dinv, N);
    k_degcount<<<(E + B - 1) / B, B, 0, stream>>>(dstp, dinv, E);
    k_rsqrt_inplace<<<(N + B - 1) / B, B, 0, stream>>>(dinv, N);

    const int gemm_blocks = ((N / 16) * 32 + B - 1) / B;    // 782
    const int rowg_blocks = (N * 16 + B - 1) / B;           // 16 threads / row
    const int edge_blocks = (E * 16 + B - 1) / B;           // 16 threads / edge
    const int sm_blocks   = (N * 32 + B - 1) / B;           // 1 wave / row

    const float* cur = x;
    float* bufs[2] = {H0, H1};
    for (int l = 0; l < 6; ++l) {
        if (l == 0)
            k_gemm_wmma<128><<<gemm_blocks, B, 0, stream>>>(cur, Ws[l], T, N);
        else
            k_gemm_wmma<64><<<gemm_blocks, B, 0, stream>>>(cur, Ws[l], T, N);
        float* nxt = bufs[l & 1];
        k_selfloop_init<<<rowg_blocks, B, 0, stream>>>(T, dinv, nxt, N);
        k_edge_scatter<<<edge_blocks, B, 0, stream>>>(srcp, dstp, dinv, T, nxt, E);
        k_bias_relu<<<rowg_blocks, B, 0, stream>>>(nxt, bs[l], N);
        cur = nxt;
    }

    k_log_softmax<<<sm_blocks, B, 0, stream>>>(cur, (float*)d_out, N);
}